// QBatchNorm2d_65807488910046
// MI455X (gfx1250) — compile-verified
//
#include <hip/hip_runtime.h>

typedef float v2f __attribute__((ext_vector_type(2)));
typedef float v8f __attribute__((ext_vector_type(8)));

#define HW    3136     // 56*56
#define BATCH 32
#define C4    256
#define CQ    64       // channels per component
#define LSTR  132      // LDS row pitch in dwords (132 % 64 == 4 -> conflict-free)
#define EPSV  1e-5f

// ---------------------------------------------------------------------------
// Kernel 1: per (channel-group, batch) partial raw moments.
//   Grid: (16 channel groups, 32 batches), 256 threads = 8 waves.
//   Each block: 16 "rows" (4 channels x 4 components) x 3136 pixels.
//   Gram (raw 2nd moments) via V_WMMA_F32_16X16X4_F32 with A == B^T aliasing;
//   row sums (for the mean) accumulated during the coalesced staging loads.
// ---------------------------------------------------------------------------
__global__ __launch_bounds__(256) void qbn_moments(const float* __restrict__ x,
                                                   float* __restrict__ wsum,
                                                   float* __restrict__ wcov)
{
    __shared__ __align__(16) float tile[16 * LSTR];   // 16 rows x 128 pixels (padded)
    __shared__ float accbuf[8 * 256];                 // per-wave 16x16 accumulators
    __shared__ float rsum[16];

    const int cg    = blockIdx.x;          // channel group 0..15 (4 channels each)
    const int batch = blockIdx.y;          // 0..31
    const int tid   = threadIdx.x;
    const int w     = tid >> 5;            // wave 0..7
    const int lane  = tid & 31;

    // staging rows for this wave
    const int m0 = 2 * w, m1 = 2 * w + 1;
    const int c0 = cg << 2;

    const float* p0 = x + ((size_t)(batch * C4 + (m0 & 3) * CQ + c0 + (m0 >> 2))) * HW;
    const float* p1 = x + ((size_t)(batch * C4 + (m1 & 3) * CQ + c0 + (m1 >> 2))) * HW;

    v8f acc = {0.f, 0.f, 0.f, 0.f, 0.f, 0.f, 0.f, 0.f};
    float rs0 = 0.f, rs1 = 0.f;

    const int mm = lane & 15;              // WMMA A-layout row
    const int kh = lane >> 4;              // K half: lanes 16-31 hold K=2,3

    for (int t = 0; t < 25; ++t) {         // 25 tiles of 128 pixels (last half-full)
        const int pix = t * 128 + (lane << 2);
        float4 v0 = make_float4(0.f, 0.f, 0.f, 0.f);
        float4 v1 = make_float4(0.f, 0.f, 0.f, 0.f);
        if (pix < HW) {                    // tail lanes pad with zeros (harmless for raw sums)
            v0 = *reinterpret_cast<const float4*>(p0 + pix);
            v1 = *reinterpret_cast<const float4*>(p1 + pix);
        }
        *reinterpret_cast<float4*>(&tile[m0 * LSTR + (lane << 2)]) = v0;
        *reinterpret_cast<float4*>(&tile[m1 * LSTR + (lane << 2)]) = v1;
        rs0 += v0.x + v0.y + v0.z + v0.w;
        rs1 += v1.x + v1.y + v1.z + v1.w;
        __syncthreads();

        // 32 K-chunks of 4 pixels per tile; wave w takes chunks {w, w+8, w+16, w+24}
        #pragma unroll
        for (int i = 0; i < 4; ++i) {
            const int q = (w + (i << 3)) << 2;     // pixel offset within tile
            v2f a = *reinterpret_cast<const v2f*>(&tile[mm * LSTR + q + (kh << 1)]);
            // D = A * A^T + C : same registers serve as A (16x4) and B (4x16)
            acc = __builtin_amdgcn_wmma_f32_16x16x4_f32(
                      false, a, false, a, (short)0, acc, false, false);
        }
        __syncthreads();
    }

    // wave32 butterfly reduce of the per-row sums
    #pragma unroll
    for (int off = 16; off > 0; off >>= 1) {
        rs0 += __shfl_xor(rs0, off, 32);
        rs1 += __shfl_xor(rs1, off, 32);
    }
    if (lane == 0) { rsum[m0] = rs0; rsum[m1] = rs1; }

    // C/D layout: VGPR v, lane L -> M = v + 8*(L>=16), N = L%16
    #pragma unroll
    for (int v = 0; v < 8; ++v)
        accbuf[(w << 8) + ((v + (kh << 3)) << 4) + mm] = acc[v];
    __syncthreads();

    const size_t bo = (size_t)(cg * BATCH + batch);
    if (tid < 16) wsum[(bo << 4) + tid] = rsum[tid];
    float s = 0.f;
    #pragma unroll
    for (int ww = 0; ww < 8; ++ww) s += accbuf[(ww << 8) + tid];
    wcov[(bo << 8) + tid] = s;
}

// ---------------------------------------------------------------------------
// Kernel 2: one block, one thread per channel.
//   Reduce partials -> mean, cov; Cholesky; invert L; fuse with affine:
//   A_c = W_c * L^-1, beta_c = b_c - A_c * mean. Stores 20 floats/channel.
// ---------------------------------------------------------------------------
__global__ __launch_bounds__(64) void qbn_solve(const float* __restrict__ wsum,
                                                const float* __restrict__ wcov,
                                                const float* __restrict__ weight,
                                                const float* __restrict__ bias,
                                                float* __restrict__ fin)
{
    const int c = threadIdx.x;
    if (c >= CQ) return;
    const int cg = c >> 2;
    const int r0 = (c & 3) << 2;
    const float Nf = (float)(BATCH * HW);

    float s[4] = {0.f, 0.f, 0.f, 0.f};
    float S[4][4] = {};
    for (int blk = 0; blk < BATCH; ++blk) {
        const float* ps = wsum + ((size_t)(cg * BATCH + blk) << 4);
        const float* pc = wcov + ((size_t)(cg * BATCH + blk) << 8);
        #pragma unroll
        for (int i = 0; i < 4; ++i) {
            s[i] += ps[r0 + i];
            #pragma unroll
            for (int j = 0; j < 4; ++j)
                S[i][j] += pc[(r0 + i) * 16 + (r0 + j)];
        }
    }

    float mean[4], cov[4][4];
    #pragma unroll
    for (int i = 0; i < 4; ++i) mean[i] = s[i] / Nf;
    for (int i = 0; i < 4; ++i)
        for (int j = 0; j < 4; ++j)
            cov[i][j] = S[i][j] / Nf - mean[i] * mean[j];
    for (int i = 0; i < 4; ++i) cov[i][i] += EPSV;

    // Cholesky (lower)
    float L[4][4] = {};
    for (int i = 0; i < 4; ++i)
        for (int j = 0; j <= i; ++j) {
            float a = cov[i][j];
            for (int k = 0; k < j; ++k) a -= L[i][k] * L[j][k];
            L[i][j] = (i == j) ? sqrtf(a) : a / L[j][j];
        }

    // invert lower-triangular L
    float Li[4][4] = {};
    for (int j = 0; j < 4; ++j) {
        Li[j][j] = 1.f / L[j][j];
        for (int i = j + 1; i < 4; ++i) {
            float a = 0.f;
            for (int k = j; k < i; ++k) a += L[i][k] * Li[k][j];
            Li[i][j] = -a / L[i][i];
        }
    }

    // A = W_c * L^-1 ; beta = b_c - A * mean
    float* F = fin + c * 20;
    for (int i = 0; i < 4; ++i)
        for (int j = 0; j < 4; ++j) {
            float a = 0.f;
            for (int k = j; k < 4; ++k) a += weight[(i * 4 + k) * CQ + c] * Li[k][j];
            F[i * 4 + j] = a;
        }
    for (int i = 0; i < 4; ++i) {
        float b = bias[i * CQ + c];
        for (int j = 0; j < 4; ++j) b -= F[i * 4 + j] * mean[j];
        F[16 + i] = b;
    }
}

// ---------------------------------------------------------------------------
// Kernel 3: fused whiten + affine as one 4x4 per-pixel transform.
//   One thread = 4 consecutive pixels (float4 in / float4 out per component).
//   Pure streaming pass: 103 MB read + 103 MB write, fully coalesced.
// ---------------------------------------------------------------------------
__global__ __launch_bounds__(256) void qbn_apply(const float* __restrict__ x,
                                                 const float* __restrict__ fin,
                                                 float* __restrict__ out)
{
    const int QHW = HW / 4;                          // 784 float4 groups per plane
    const int idx = blockIdx.x * 256 + threadIdx.x;  // < 64*32*784, exact grid
    const int c   = idx / (BATCH * QHW);
    const int rem = idx - c * (BATCH * QHW);
    const int b   = rem / QHW;
    const int q   = rem - b * QHW;

    const float* F = fin + c * 20;
    float A[16], beta[4];
    #pragma unroll
    for (int i = 0; i < 16; ++i) A[i] = F[i];
    #pragma unroll
    for (int i = 0; i < 4; ++i) beta[i] = F[16 + i];

    const size_t base = ((size_t)(b * C4 + c)) * HW + (q << 2);
    const size_t cs   = (size_t)CQ * HW;
    const float4 x0 = *reinterpret_cast<const float4*>(x + base);
    const float4 x1 = *reinterpret_cast<const float4*>(x + base + cs);
    const float4 x2 = *reinterpret_cast<const float4*>(x + base + 2 * cs);
    const float4 x3 = *reinterpret_cast<const float4*>(x + base + 3 * cs);

    #pragma unroll
    for (int i = 0; i < 4; ++i) {
        float4 o;
        o.x = A[i*4+0]*x0.x + A[i*4+1]*x1.x + A[i*4+2]*x2.x + A[i*4+3]*x3.x + beta[i];
        o.y = A[i*4+0]*x0.y + A[i*4+1]*x1.y + A[i*4+2]*x2.y + A[i*4+3]*x3.y + beta[i];
        o.z = A[i*4+0]*x0.z + A[i*4+1]*x1.z + A[i*4+2]*x2.z + A[i*4+3]*x3.z + beta[i];
        o.w = A[i*4+0]*x0.w + A[i*4+1]*x1.w + A[i*4+2]*x2.w + A[i*4+3]*x3.w + beta[i];
        *reinterpret_cast<float4*>(out + ((size_t)(b * C4 + i * CQ + c)) * HW + (q << 2)) = o;
    }
}

// ---------------------------------------------------------------------------
extern "C" void kernel_launch(void* const* d_in, const int* in_sizes, int n_in,
                              void* d_out, int out_size, void* d_ws, size_t ws_size,
                              hipStream_t stream)
{
    const float* x      = (const float*)d_in[0];   // [32, 256, 56, 56]
    const float* weight = (const float*)d_in[1];   // [4, 4, 64]
    const float* bias   = (const float*)d_in[2];   // [4, 64]
    float* out = (float*)d_out;

    // workspace layout (floats):
    //   wsum: 512 blocks x 16 rows          =   8192
    //   wcov: 512 blocks x 256 (16x16 Gram) = 131072
    //   fin : 64 channels x 20 (A|beta)     =   1280
    float* wsum = (float*)d_ws;
    float* wcov = wsum + 512 * 16;
    float* fin  = wcov + 512 * 256;

    qbn_moments<<<dim3(16, BATCH), 256, 0, stream>>>(x, wsum, wcov);
    qbn_solve<<<1, 64, 0, stream>>>(wsum, wcov, weight, bias, fin);
    qbn_apply<<<(CQ * BATCH * (HW / 4)) / 256, 256, 0, stream>>>(x, fin, out);
}